// RipsECC_76879914598458
// MI455X (gfx1250) — compile-verified
//
#include <hip/hip_runtime.h>

typedef float v2f __attribute__((ext_vector_type(2)));
typedef float v8f __attribute__((ext_vector_type(8)));

#define NPTS    4096
#define STEPS   64
#define BSCALE  31.5f                    // (STEPS-1)/(T_MAX-T_MIN) = 63/2
#define STILE   32                       // super-tile = 2x2 WMMA tiles
#define NST     (NPTS / STILE)           // 128
#define NPAIRS  (NST * (NST + 1) / 2)    // 8256 upper-tri super-tile pairs
#define WPB     8                        // waves per block (256 thr, wave32)
#define HBINS   80                       // 64 real + dummy(64) + bank stagger
#define DUMMYB  64

static __device__ __forceinline__ v8f wmma_f32_4(v2f a, v2f b) {
    v8f z = {0.f, 0.f, 0.f, 0.f, 0.f, 0.f, 0.f, 0.f};
    return __builtin_amdgcn_wmma_f32_16x16x4_f32(
        false, a, false, b, (short)0, z, false, false);
}

// ---------------------------------------------- zero hist + point norms
__global__ void ecc_init_kernel(const float* __restrict__ x,
                                unsigned* __restrict__ ghist,
                                float* __restrict__ nrm) {
    const int i = blockIdx.x * 256 + threadIdx.x;
    if (i < NPTS) {
        const float a = x[i * 3 + 0], b = x[i * 3 + 1], c = x[i * 3 + 2];
        nrm[i] = a * a + b * b + c * c;
    }
    if (blockIdx.x == 0 && threadIdx.x < STEPS) ghist[threadIdx.x] = 0u;
}

// ------------------------------------------ 32x32 super-tiles, 4 WMMAs/wave
__global__ void __launch_bounds__(256)
ecc_pairs_kernel(const float* __restrict__ x, const float* __restrict__ nrm,
                 unsigned* __restrict__ ghist) {
    __shared__ unsigned hist[WPB][HBINS];   // per-wave sub-histograms

    const int tid  = threadIdx.x;
    const int wave = tid >> 5;
    const int lane = tid & 31;
    const int m    = lane & 15;             // row for A/B operands, N for D
    const int half = lane >> 4;             // K-half for A/B, +8 M rows for D

    for (int b = tid; b < WPB * HBINS; b += 256) ((unsigned*)hist)[b] = 0u;
    __syncthreads();

    const int k = blockIdx.x * WPB + wave;  // flat upper-tri super-tile index
    if (k < NPAIRS) {
        // decode k -> (ti, tj), ti <= tj, over a 128x128 super-tile grid
        int ti = (int)((257.0f - __builtin_amdgcn_sqrtf(
                            66049.0f - 8.0f * (float)k)) * 0.5f);
        ti = ti < 0 ? 0 : (ti > NST - 1 ? NST - 1 : ti);
        while (ti > 0 && (ti * NST - (ti * (ti - 1)) / 2) > k) --ti;
        while (((ti + 1) * NST - ((ti + 1) * ti) / 2) <= k) ++ti;
        const int tj = ti + (k - (ti * NST - (ti * (ti - 1)) / 2));

        const int gi0 = ti * STILE;
        const int gj0 = tj * STILE;

        // A (16x4 f32, K=3 zero-padded) and B (4x16) share the same per-lane
        // pattern: lanes 0-15 carry K={0,1}, lanes 16-31 carry K={2,pad}.
        // cndmask selects keep EXEC all-1s for the WMMAs.
        auto mk_op = [&](const float* p) -> v2f {
            v2f r;
            r.x = half ? p[2] : p[0];
            r.y = half ? 0.0f : p[1];
            return r;
        };
        const v2f a0 = mk_op(x + (gi0 + m) * 3);
        const v2f a1 = mk_op(x + (gi0 + 16 + m) * 3);
        const v2f b0 = mk_op(x + (gj0 + m) * 3);
        const v2f b1 = mk_op(x + (gj0 + 16 + m) * 3);

        // 2x2 block of 16x16 dot-product tiles (independent accumulators)
        const v8f d00 = wmma_f32_4(a0, b0);
        const v8f d01 = wmma_f32_4(a0, b1);
        const v8f d10 = wmma_f32_4(a1, b0);
        const v8f d11 = wmma_f32_4(a1, b1);

        const float nj0 = nrm[gj0 + m];
        const float nj1 = nrm[gj0 + 16 + m];
        const int   j0  = gj0 + m;
        const int   j1  = gj0 + 16 + m;

        unsigned* h = hist[wave];
        #pragma unroll
        for (int v = 0; v < 8; ++v) {
            #pragma unroll
            for (int ri = 0; ri < 2; ++ri) {
                const int   i  = gi0 + ri * 16 + half * 8 + v;
                const float ni = nrm[i];
                const float dA = ri ? d10[v] : d00[v];
                const float dB = ri ? d11[v] : d01[v];
                {   // column block 0
                    const float d2   = fmaxf(fmaf(-2.0f, dA, ni + nj0), 0.0f);
                    const float dist = __builtin_amdgcn_sqrtf(d2);  // raw v_sqrt_f32
                    int bin = (int)__builtin_ceilf(dist * BSCALE);
                    bin = bin < 0 ? 0 : (bin > STEPS - 1 ? STEPS - 1 : bin);
                    bin = (i < j0 && dist <= 2.0f) ? bin : DUMMYB;  // branchless mask
                    atomicAdd(&h[bin], 1u);                          // ds_add_u32
                }
                {   // column block 1
                    const float d2   = fmaxf(fmaf(-2.0f, dB, ni + nj1), 0.0f);
                    const float dist = __builtin_amdgcn_sqrtf(d2);
                    int bin = (int)__builtin_ceilf(dist * BSCALE);
                    bin = bin < 0 ? 0 : (bin > STEPS - 1 ? STEPS - 1 : bin);
                    bin = (i < j1 && dist <= 2.0f) ? bin : DUMMYB;
                    atomicAdd(&h[bin], 1u);
                }
            }
        }
    }

    __syncthreads();
    // reduce per-wave sub-histograms (dummy bin 64 discarded) -> global
    if (tid < STEPS) {
        unsigned s = 0u;
        #pragma unroll
        for (int w = 0; w < WPB; ++w) s += hist[w][tid];
        if (s) atomicAdd(&ghist[tid], s);            // global_atomic_add_u32
    }
}

// ------------------------------------------------ vertices + cumsum -> out
__global__ void ecc_final_kernel(const unsigned* __restrict__ ghist,
                                 float* __restrict__ out) {
    if (threadIdx.x == 0 && blockIdx.x == 0) {
        float acc = 0.0f;
        for (int b = 0; b < STEPS; ++b) {
            const float e = (b == 0 ? (float)NPTS : 0.0f) - (float)ghist[b];
            acc += e;
            out[b] = acc;
        }
    }
}

extern "C" void kernel_launch(void* const* d_in, const int* in_sizes, int n_in,
                              void* d_out, int out_size, void* d_ws, size_t ws_size,
                              hipStream_t stream) {
    const float* x    = (const float*)d_in[0];                 // [4096, 3] f32
    float*       out  = (float*)d_out;                         // [64] f32
    unsigned*    hist = (unsigned*)d_ws;                       // 64 u32 bins
    float*       nrm  = (float*)((char*)d_ws + 256);           // 4096 f32 norms

    ecc_init_kernel<<<(NPTS + 255) / 256, 256, 0, stream>>>(x, hist, nrm);

    const int nblocks = (NPAIRS + WPB - 1) / WPB;              // 1032
    ecc_pairs_kernel<<<nblocks, 256, 0, stream>>>(x, nrm, hist);

    ecc_final_kernel<<<1, 32, 0, stream>>>(hist, out);
}